// HadamardTransform_3728031613740
// MI455X (gfx1250) — compile-verified
//
#include <hip/hip_runtime.h>

// out = (1/sqrt(2048)) * value @ H2048, H2048 = H8 (x) H16 (x) H16 (Sylvester, symmetric).
// One wave (32 lanes) transforms one 2048-float row entirely in LDS using
// fp32 WMMA (V_WMMA_F32_16X16X4_F32) for the two 16-point stages and VALU
// butterflies for the 8-point stage. Global<->LDS movement uses the CDNA5
// async-to-LDS path (ASYNCcnt). Bandwidth-bound: 512 MB total traffic.

typedef float v2f __attribute__((ext_vector_type(2)));
typedef float v8f __attribute__((ext_vector_type(8)));
typedef __attribute__((address_space(3))) float lds_float;

#define HN    2048
#define WAVES 8
#define HAD_SCALE 0.02209708691207961f /* 1/sqrt(2048) */

// INST_OFFSET is added to BOTH the LDS and the global address (ISA 08 §4.4),
// so one base pair + immediate offsets walks the whole row.
#define ASYNC_LOAD_B128(off)                                                   \
  asm volatile("global_load_async_to_lds_b128 %0, %1, off offset:" #off       \
               :: "v"(lbase), "v"(gsrc) : "memory")
#define ASYNC_STORE_B128(off)                                                  \
  asm volatile("global_store_async_from_lds_b128 %0, %1, off offset:" #off    \
               :: "v"(gdst), "v"(lbase) : "memory")

__device__ __forceinline__ v8f wmma4(v2f a, v2f b, v8f c) {
  // 8 args: (neg_a, A, neg_b, B, c_mod, C, reuse_a, reuse_b)
  return __builtin_amdgcn_wmma_f32_16x16x4_f32(false, a, false, b, (short)0, c,
                                               false, false);
}

__global__ __launch_bounds__(256)
void hadamard2048_wmma_kernel(const float* __restrict__ x,
                              float* __restrict__ y, int nrows) {
  __shared__ float lds[WAVES][HN];

  const int lane = threadIdx.x & 31;
  const int wave = threadIdx.x >> 5;          // wave32
  const int row  = blockIdx.x * WAVES + wave; // wave-uniform -> EXEC stays all-1s
  if (row >= nrows) return;

  const float* __restrict__ src = x + (size_t)row * HN;
  float*       __restrict__ dst = y + (size_t)row * HN;
  float* buf = lds[wave];

  // ---- async row load: global -> LDS directly (b128, ASYNCcnt) ----
  const unsigned lbase =
      (unsigned)(unsigned long long)(lds_float*)buf + (unsigned)(lane * 16);
  const unsigned long long gsrc =
      (unsigned long long)src + (unsigned long long)(lane * 16);
  ASYNC_LOAD_B128(0);    ASYNC_LOAD_B128(512);  ASYNC_LOAD_B128(1024);
  ASYNC_LOAD_B128(1536); ASYNC_LOAD_B128(2048); ASYNC_LOAD_B128(2560);
  ASYNC_LOAD_B128(3072); ASYNC_LOAD_B128(3584); ASYNC_LOAD_B128(4096);
  ASYNC_LOAD_B128(4608); ASYNC_LOAD_B128(5120); ASYNC_LOAD_B128(5632);
  ASYNC_LOAD_B128(6144); ASYNC_LOAD_B128(6656); ASYNC_LOAD_B128(7168);
  ASYNC_LOAD_B128(7680);
  asm volatile("s_wait_asynccnt 0x0" ::: "memory"); // row resident in LDS

  const int m    = lane & 15;  // M (A) / N (B,C,D) index of this lane
  const int half = lane >> 4;  // selects K pair within a 4-wide chunk

  // ---- H16 as B operand, built once in registers ----
  // chunk q covers K rows [4q, 4q+4); within chunk: K = 4q + 2*half + v
  // B[k][n] = (-1)^popc(k & n)
  v2f B[4];
  #pragma unroll
  for (int q = 0; q < 4; ++q) {
    int k = 4 * q + 2 * half;
    B[q].x = (__popc(k & m) & 1) ? -1.0f : 1.0f;
    B[q].y = (__popc((k + 1) & m) & 1) ? -1.0f : 1.0f;
  }

  // ---- Stage 1: y[k1,k0,j0] = sum_i0 x[k1,k0,i0] * H16[i0,j0] ----
  // tile t == k1; A[mm][k] = buf[256t + 16*mm + k]  (row-major 16x16 block)
  #pragma unroll
  for (int t = 0; t < 8; ++t) {
    v8f acc = {0.f, 0.f, 0.f, 0.f, 0.f, 0.f, 0.f, 0.f};
    #pragma unroll
    for (int q = 0; q < 4; ++q) {
      v2f a = *(const v2f*)(buf + 256 * t + 16 * m + 4 * q + 2 * half);
      acc = wmma4(a, B[q], acc);
    }
    // D[mm][n]: lane half selects mm = r (+8); write y back in place: [t][k0][j0]
    #pragma unroll
    for (int r = 0; r < 8; ++r)
      buf[256 * t + (r + 8 * half) * 16 + m] = acc[r];
  }

  // ---- Stage 2: w[k1,m0,j0] = sum_k0 H16[m0,k0] * y[k1,k0,j0] ----
  // H16 symmetric => compute (y_t^T) @ H16; A[mm=j0][k=k0] = buf[256t + k*16 + mm]
  #pragma unroll
  for (int t = 0; t < 8; ++t) {
    v8f acc = {0.f, 0.f, 0.f, 0.f, 0.f, 0.f, 0.f, 0.f};
    #pragma unroll
    for (int q = 0; q < 4; ++q) {
      int k = 4 * q + 2 * half;
      v2f a;
      a.x = buf[256 * t + k * 16 + m];
      a.y = buf[256 * t + (k + 1) * 16 + m];
      acc = wmma4(a, B[q], acc);
    }
    // in-place store: [t][j0][m0]  (w transposed within the plane)
    #pragma unroll
    for (int r = 0; r < 8; ++r)
      buf[256 * t + (r + 8 * half) * 16 + m] = acc[r];
  }

  // ---- Stage 3: FWHT-8 across tile planes (k1) + in-plane transpose + scale ----
  // Load EVERYTHING first so the in-place transposed stores are hazard-free
  // (per-wave DS ops complete in order).
  float v[8][8];
  #pragma unroll
  for (int i = 0; i < 8; ++i) {
    int p = i * 32 + lane; // p = j0*16 + m0 within a plane
    #pragma unroll
    for (int t = 0; t < 8; ++t) v[i][t] = buf[t * 256 + p];
  }
  #pragma unroll
  for (int i = 0; i < 8; ++i) {
    int p  = i * 32 + lane;
    int j0 = p >> 4;
    int m0 = p & 15;
    float a0 = v[i][0] + v[i][1], a1 = v[i][0] - v[i][1];
    float a2 = v[i][2] + v[i][3], a3 = v[i][2] - v[i][3];
    float a4 = v[i][4] + v[i][5], a5 = v[i][4] - v[i][5];
    float a6 = v[i][6] + v[i][7], a7 = v[i][6] - v[i][7];
    float b0 = a0 + a2, b1 = a1 + a3, b2 = a0 - a2, b3 = a1 - a3;
    float b4 = a4 + a6, b5 = a5 + a7, b6 = a4 - a6, b7 = a5 - a7;
    float o0 = b0 + b4, o1 = b1 + b5, o2 = b2 + b6, o3 = b3 + b7;
    float o4 = b0 - b4, o5 = b1 - b5, o6 = b2 - b6, o7 = b3 - b7;
    // final layout: out[n'], n' = m1*256 + m0*16 + j0
    int q = m0 * 16 + j0;
    buf[0 * 256 + q] = o0 * HAD_SCALE;
    buf[1 * 256 + q] = o1 * HAD_SCALE;
    buf[2 * 256 + q] = o2 * HAD_SCALE;
    buf[3 * 256 + q] = o3 * HAD_SCALE;
    buf[4 * 256 + q] = o4 * HAD_SCALE;
    buf[5 * 256 + q] = o5 * HAD_SCALE;
    buf[6 * 256 + q] = o6 * HAD_SCALE;
    buf[7 * 256 + q] = o7 * HAD_SCALE;
  }

  // ---- async row store: LDS -> global directly (b128, ASYNCcnt) ----
  // DS writes and async LDS reads are NOT hardware-ordered: drain DScnt first.
  asm volatile("s_wait_dscnt 0x0" ::: "memory");
  const unsigned long long gdst =
      (unsigned long long)dst + (unsigned long long)(lane * 16);
  ASYNC_STORE_B128(0);    ASYNC_STORE_B128(512);  ASYNC_STORE_B128(1024);
  ASYNC_STORE_B128(1536); ASYNC_STORE_B128(2048); ASYNC_STORE_B128(2560);
  ASYNC_STORE_B128(3072); ASYNC_STORE_B128(3584); ASYNC_STORE_B128(4096);
  ASYNC_STORE_B128(4608); ASYNC_STORE_B128(5120); ASYNC_STORE_B128(5632);
  ASYNC_STORE_B128(6144); ASYNC_STORE_B128(6656); ASYNC_STORE_B128(7168);
  ASYNC_STORE_B128(7680);
  asm volatile("s_wait_asynccnt 0x0" ::: "memory");
}

extern "C" void kernel_launch(void* const* d_in, const int* in_sizes, int n_in,
                              void* d_out, int out_size, void* d_ws, size_t ws_size,
                              hipStream_t stream) {
  const float* value = (const float*)d_in[0]; // (nrows, 2048) fp32
  // d_in[1] (the Hadamard weight) is structured: not needed.
  float* out = (float*)d_out;
  int nrows  = in_sizes[0] / HN;
  int blocks = (nrows + WAVES - 1) / WAVES;
  hipLaunchKernelGGL(hadamard2048_wmma_kernel, dim3(blocks), dim3(256), 0,
                     stream, value, out, nrows);
}